// CensoredCrossEntropyLoss_11940009082937
// MI455X (gfx1250) — compile-verified
//
#include <hip/hip_runtime.h>

typedef float v2f __attribute__((ext_vector_type(2)));
typedef float v8f __attribute__((ext_vector_type(8)));

#define WAVES   8
#define BLOCK   256
#define NBLK    2048
#define LOG2E   1.44269504088896340736f
#define LN2     0.69314718055994530942f
#define LOG_EPS -27.631021115928547f   /* ln(1e-12) */

// ---------------------------------------------------------------------------
// Kernel 1: one wave32 processes 16 rows (4KB tile). Suffix-cumsum is done as
// a triangular GEMM on the matrix unit: suf = E x T, T[k][j] = (k >= j),
// via chained V_WMMA_F32_16X16X4_F32 (full fp32 accuracy).
// ---------------------------------------------------------------------------
__global__ __launch_bounds__(BLOCK) void cce_partial_kernel(
    const float* __restrict__ x, const long long* __restrict__ y,
    const float* __restrict__ obs, float* __restrict__ part, int numTiles)
{
    __shared__ float lds[WAVES][16][68];   // padded stride 68 -> conflict-free
    __shared__ float wsum[WAVES];

    const int w    = threadIdx.x >> 5;
    const int lane = threadIdx.x & 31;
    const int r    = lane & 15;            // row within tile (A-matrix M)
    const int h    = lane >> 4;            // half-wave selects K pairs
    float (*buf)[68] = lds[w];

    const int wid = blockIdx.x * WAVES + w;
    const int nw  = NBLK * WAVES;

    float acc_loss = 0.0f;

    for (int tile = wid; tile < numTiles; tile += nw) {
        const float* xt = x + (size_t)tile * 1024u;

        // Load 32 values/lane directly in WMMA A-layout:
        // lane L(<16): row L, K = 4c+{0,1}; lane L+16: row L, K = 4c+{2,3}
        v2f xa[16];
        #pragma unroll
        for (int c = 0; c < 16; ++c) {
            int col = 4 * c + 2 * h;
            xa[c] = *(const v2f*)(xt + r * 64 + col);
        }
        // Stage raw x into LDS (only needed to fetch x[row][y] later)
        #pragma unroll
        for (int c = 0; c < 16; ++c) {
            int col = 4 * c + 2 * h;
            *(v2f*)&buf[r][col] = xa[c];
        }
        asm volatile("s_wait_dscnt 0" ::: "memory");

        int   yv = 0;
        float ob = 0.0f, xy = 0.0f;
        if (lane < 16) {
            int row = tile * 16 + r;
            yv = (int)y[row];              // int64 label, low word suffices
            ob = obs[row];
            xy = buf[r][yv];
        }
        asm volatile("s_wait_dscnt 0" ::: "memory");

        // Row max: per-lane tree + cross-half swap
        float m = fmaxf(xa[0].x, xa[0].y);
        #pragma unroll
        for (int c = 1; c < 16; ++c)
            m = fmaxf(m, fmaxf(xa[c].x, xa[c].y));
        m = fmaxf(m, __shfl_xor(m, 16, 32));

        // e = exp(x - m)  (v_exp_f32 path)
        #pragma unroll
        for (int c = 0; c < 16; ++c) {
            xa[c].x = __builtin_amdgcn_exp2f((xa[c].x - m) * LOG2E);
            xa[c].y = __builtin_amdgcn_exp2f((xa[c].y - m) * LOG2E);
        }

        // suf[i][j] = sum_{k>=j} e[i][k] = (E x T)[i][j], 40 WMMAs per tile
        const int colbase = lane & 15;
        #pragma unroll
        for (int t = 0; t < 4; ++t) {
            v8f acc = {0.f, 0.f, 0.f, 0.f, 0.f, 0.f, 0.f, 0.f};
            #pragma unroll
            for (int c = 4 * t; c < 16; ++c) {   // c < 4t chunks are all-zero B
                int k0 = 4 * c + 2 * h;
                int j  = colbase + 16 * t;
                v2f b;
                b.x = (k0     >= j) ? 1.0f : 0.0f;
                b.y = (k0 + 1 >= j) ? 1.0f : 0.0f;
                acc = __builtin_amdgcn_wmma_f32_16x16x4_f32(
                          false, xa[c], false, b, (short)0, acc, false, false);
            }
            // D layout: VGPR rr -> row (8h + rr), column = colbase + 16t
            #pragma unroll
            for (int rr = 0; rr < 8; ++rr)
                buf[h * 8 + rr][colbase + 16 * t] = acc[rr];
        }
        asm volatile("s_wait_dscnt 0" ::: "memory");

        if (lane < 16) {
            float S   = buf[r][0];                 // full row sum of e
            int   i1  = (yv < 63) ? (yv + 1) : 63;
            float s1  = buf[r][i1];
            float lnS = __builtin_amdgcn_logf(S) * LN2;
            float term;
            if (ob > 0.5f || yv == 63) {
                // log p_y = (x_y - m) - ln S, clamped at ln(EPS)
                term = fmaxf((xy - m) - lnS, LOG_EPS);
            } else {
                float lnsuf = __builtin_amdgcn_logf(s1) * LN2;  // -inf ok if 0
                term = fmaxf(lnsuf - lnS, LOG_EPS);
            }
            acc_loss += term;
        }
        asm volatile("s_wait_dscnt 0" ::: "memory");
    }

    // Deterministic reduction: wave shfl tree -> per-block partial
    float v = acc_loss;
    v += __shfl_xor(v, 16, 32);
    v += __shfl_xor(v, 8, 32);
    v += __shfl_xor(v, 4, 32);
    v += __shfl_xor(v, 2, 32);
    v += __shfl_xor(v, 1, 32);
    if (lane == 0) wsum[w] = v;
    __syncthreads();
    if (threadIdx.x == 0) {
        float b = 0.0f;
        #pragma unroll
        for (int i = 0; i < WAVES; ++i) b += wsum[i];
        part[blockIdx.x] = b;
    }
}

// ---------------------------------------------------------------------------
// Kernel 2: scalar handling of the < 16 remainder rows (N % 16)
// ---------------------------------------------------------------------------
__global__ __launch_bounds__(32) void cce_tail_kernel(
    const float* __restrict__ x, const long long* __restrict__ y,
    const float* __restrict__ obs, float* __restrict__ part, int N0, int N)
{
    __shared__ float s[32];
    int t = threadIdx.x;
    float term = 0.0f;
    int row = N0 + t;
    if (row < N) {
        const float* xr = x + (size_t)row * 64u;
        float m = xr[0];
        for (int k = 1; k < 64; ++k) m = fmaxf(m, xr[k]);
        int yv = (int)y[row];
        float xy = xr[yv];
        float S = 0.0f, suf = 0.0f;
        for (int k = 0; k < 64; ++k) {
            float e = __builtin_amdgcn_exp2f((xr[k] - m) * LOG2E);
            S += e;
            if (k >= yv + 1) suf += e;
        }
        float lnS = __builtin_amdgcn_logf(S) * LN2;
        if (obs[row] > 0.5f || yv == 63)
            term = fmaxf((xy - m) - lnS, LOG_EPS);
        else
            term = fmaxf(__builtin_amdgcn_logf(suf) * LN2 - lnS, LOG_EPS);
    }
    s[t] = term;
    __syncthreads();
    if (t == 0) {
        float b = 0.0f;
        for (int i = 0; i < 32; ++i) b += s[i];
        part[NBLK] = b;
    }
}

// ---------------------------------------------------------------------------
// Kernel 3: deterministic final reduction of NBLK+1 partials -> -mean
// ---------------------------------------------------------------------------
__global__ __launch_bounds__(256) void cce_final_kernel(
    const float* __restrict__ part, int P, float* __restrict__ out, float invN)
{
    __shared__ float s[256];
    float a = 0.0f;
    for (int i = threadIdx.x; i < P; i += 256) a += part[i];
    s[threadIdx.x] = a;
    __syncthreads();
    for (int st = 128; st > 0; st >>= 1) {
        if (threadIdx.x < st) s[threadIdx.x] += s[threadIdx.x + st];
        __syncthreads();
    }
    if (threadIdx.x == 0) out[0] = -s[0] * invN;
}

extern "C" void kernel_launch(void* const* d_in, const int* in_sizes, int n_in,
                              void* d_out, int out_size, void* d_ws, size_t ws_size,
                              hipStream_t stream)
{
    const float*     x   = (const float*)d_in[0];
    const long long* y   = (const long long*)d_in[1];
    const float*     obs = (const float*)d_in[2];
    float* out  = (float*)d_out;
    float* part = (float*)d_ws;   // NBLK+1 floats of scratch

    const int N        = in_sizes[1];     // y has N elements
    const int numTiles = N >> 4;
    const int N0       = numTiles << 4;

    cce_partial_kernel<<<NBLK, BLOCK, 0, stream>>>(x, y, obs, part, numTiles);
    cce_tail_kernel<<<1, 32, 0, stream>>>(x, y, obs, part, N0, N);
    cce_final_kernel<<<1, 256, 0, stream>>>(part, NBLK + 1, out, 1.0f / (float)N);

    (void)n_in; (void)out_size; (void)ws_size;
}